// MLP_difs_maxpool_45337674776740
// MI455X (gfx1250) — compile-verified
//
#include <hip/hip_runtime.h>

#define NEG_INF (-__builtin_inff())

// ---------------------------------------------------------------------------
// Kernel: edge-parallel scatter-max.  (Placed first in the file so the
// disasm snippet shows the atomic lowering: expect global_atomic_max_num_f32,
// no-return form, 4 per thread — NOT a cmpswap loop.)
//
//   16 lanes cooperate on one edge: lane k loads float4 #k of the 64-float
//   source row (global_load_b128; lanes 0..15 of a wave32 cover one
//   contiguous 256B row, lanes 16..31 cover the next edge's row) and folds
//   it into the destination row with 4 device-scope relaxed FP-max atomics
//   resolved in L2.  Edge indices are streamed with non-temporal loads so
//   the 12.8MB index stream does not evict the 25.6MB feature table.
// ---------------------------------------------------------------------------
__global__ __launch_bounds__(256) void k_scatter_max(const float4* __restrict__ h4,
                                                     const int* __restrict__ src,
                                                     const int* __restrict__ dst,
                                                     float* __restrict__ out,
                                                     int n_edges) {
    int t    = blockIdx.x * 256 + threadIdx.x;
    int e    = t >> 4;      // edge id
    int lane = t & 15;      // which float4 of the 64-float row
    if (e >= n_edges) return;

    int s = __builtin_nontemporal_load(&src[e]);
    int d = __builtin_nontemporal_load(&dst[e]);

    float4 v = h4[(size_t)s * 16 + lane];   // global_load_b128, L2-resident

    float* o = out + (size_t)d * 64 + (size_t)lane * 4;
    __hip_atomic_fetch_max(o + 0, v.x, __ATOMIC_RELAXED, __HIP_MEMORY_SCOPE_AGENT);
    __hip_atomic_fetch_max(o + 1, v.y, __ATOMIC_RELAXED, __HIP_MEMORY_SCOPE_AGENT);
    __hip_atomic_fetch_max(o + 2, v.z, __ATOMIC_RELAXED, __HIP_MEMORY_SCOPE_AGENT);
    __hip_atomic_fetch_max(o + 3, v.w, __ATOMIC_RELAXED, __HIP_MEMORY_SCOPE_AGENT);
}

// ---------------------------------------------------------------------------
// Kernel: fill the output with -inf (segment_max identity), 128-bit stores.
// ---------------------------------------------------------------------------
__global__ __launch_bounds__(256) void k_init_neginf(float4* __restrict__ out, int n4) {
    int i = blockIdx.x * 256 + threadIdx.x;
    if (i < n4) {
        out[i] = make_float4(NEG_INF, NEG_INF, NEG_INF, NEG_INF);
    }
}

// ---------------------------------------------------------------------------
// Kernel: nodes with no incoming edges still hold -inf; the reference maps
// those to 0.  128-bit read-modify-write.
// ---------------------------------------------------------------------------
__global__ __launch_bounds__(256) void k_finalize(float4* __restrict__ out, int n4) {
    int i = blockIdx.x * 256 + threadIdx.x;
    if (i < n4) {
        float4 v = out[i];
        v.x = (v.x == NEG_INF) ? 0.0f : v.x;
        v.y = (v.y == NEG_INF) ? 0.0f : v.y;
        v.z = (v.z == NEG_INF) ? 0.0f : v.z;
        v.w = (v.w == NEG_INF) ? 0.0f : v.w;
        out[i] = v;
    }
}

// ---------------------------------------------------------------------------
// Launch: inputs are h [N*64 f32] and edge_index [2*E i32] (row 0 = src,
// row 1 = dst, flattened). Output is [N*64] f32.
// ---------------------------------------------------------------------------
extern "C" void kernel_launch(void* const* d_in, const int* in_sizes, int n_in,
                              void* d_out, int out_size, void* d_ws, size_t ws_size,
                              hipStream_t stream) {
    (void)n_in; (void)d_ws; (void)ws_size;

    const float* h    = (const float*)d_in[0];
    const int*   edge = (const int*)d_in[1];
    const int    E    = in_sizes[1] / 2;
    const int*   src  = edge;
    const int*   dst  = edge + E;

    float* out = (float*)d_out;
    const int n4 = out_size / 4;               // number of float4's in out

    // 1) out = -inf
    k_init_neginf<<<(n4 + 255) / 256, 256, 0, stream>>>((float4*)out, n4);

    // 2) edge scatter-max: 16 threads per edge
    long long work = (long long)E * 16;
    int blocks = (int)((work + 255) / 256);
    k_scatter_max<<<blocks, 256, 0, stream>>>((const float4*)h, src, dst, out, E);

    // 3) -inf -> 0
    k_finalize<<<(n4 + 255) / 256, 256, 0, stream>>>((float4*)out, n4);
}